// S2ConvNet_5128190952048
// MI455X (gfx1250) — compile-verified
//
#include <hip/hip_runtime.h>
#include <math.h>

typedef float v2f __attribute__((ext_vector_type(2)));
typedef float v8f __attribute__((ext_vector_type(8)));

#define HD __host__ __device__
static constexpr double PI_D = 3.14159265358979323846;

// ---------------- D-table geometry (shared host/device constexpr) ----------
constexpr int NB = 6;
constexpr int DBh[NB] = {13, 10, 8, 5, 3, 2};   // bandwidths with D tables
constexpr int DLh[NB] = {10, 10, 8, 5, 3, 2};   // #l entries per bandwidth
HD constexpr long cumsq(int l) { return (long)l * (2 * l - 1) * (2 * l + 1) / 3; } // sum_{j<l}(2j+1)^2
HD constexpr long d_tab_size(int bi) { return (long)(2 * DBh[bi]) * cumsq(DLh[bi]); }
HD constexpr long d_base(int bi) { long o = 0; for (int i = 0; i < bi; i++) o += d_tab_size(i); return o; }
HD constexpr long d_off(int bi, int l) { return d_base(bi) + (long)(2 * DBh[bi]) * cumsq(l); }
constexpr long D_TOTAL = d_base(NB);            // 73960 floats

constexpr int WOFFh[NB] = {0, 26, 46, 62, 72, 78};
constexpr int W_TOTAL = 82;

// ---------------- workspace layout (float offsets) -------------------------
constexpr long PLANE  = 32768000;  // 64*64*20^3 (max complex plane)
constexpr long APL    = 1000000;   // A plane (re or im)
constexpr long BSZ    = 1500000;   // packed weight matrix
constexpr long ZPL    = 1500000;   // z plane (re or im)
constexpr long OFF_D    = 0;
constexpr long OFF_W    = OFF_D + 74000;
constexpr long OFF_REAL = OFF_W + 128;
constexpr long OFF_CA   = OFF_REAL + PLANE;        // Xf (complex, planar)
constexpr long OFF_CB   = OFF_CA + 2 * PLANE;      // out accumulator (complex)
constexpr long OFF_CC   = OFF_CB + 2 * PLANE;      // fft/ifft temp (complex)
constexpr long OFF_A    = OFF_CC + 2 * PLANE;      // GEMM A (2 planes)
constexpr long OFF_B    = OFF_A + 2 * APL;         // GEMM B (real)
constexpr long OFF_Z    = OFF_B + BSZ;             // GEMM C (2 planes)
constexpr long OFF_FCA  = OFF_Z + 2 * ZPL;
constexpr long OFF_FCB  = OFF_FCA + 40960;

// ---------------- device helpers -------------------------------------------
__device__ double dfact(int n) { double r = 1.0; for (int i = 2; i <= n; i++) r *= (double)i; return r; }
__device__ double dipow(double x, int e) { double r = 1.0; for (int i = 0; i < e; i++) r *= x; return r; }
__device__ __forceinline__ int pmod(int a, int n) { int r = a % n; return (r < 0) ? r + n : r; }

// ---------------- init: Wigner-d tables ------------------------------------
__global__ void k_init_d(float* __restrict__ D) {
  const int DBv[NB] = {13, 10, 8, 5, 3, 2};
  const int DLv[NB] = {10, 10, 8, 5, 3, 2};
  long stride = (long)gridDim.x * blockDim.x;
  for (long t = (long)blockIdx.x * blockDim.x + threadIdx.x; t < D_TOTAL; t += stride) {
    long rem = t; int b = 2, l = 0; bool found = false;
    for (int bi = 0; bi < NB && !found; bi++) {
      for (int li = 0; li < DLv[bi]; li++) {
        long sz = (long)(2 * DBv[bi]) * (2 * li + 1) * (2 * li + 1);
        if (rem < sz) { b = DBv[bi]; l = li; found = true; break; }
        rem -= sz;
      }
    }
    int n = 2 * l + 1;
    int j  = (int)(rem / (n * n));
    int r2 = (int)(rem % (n * n));
    int mi = r2 / n, ni = r2 % n;
    int m = mi - l, nn = ni - l;
    double beta = PI_D * (2.0 * j + 1.0) / (4.0 * b);
    double cb = cos(beta * 0.5), sb = sin(beta * 0.5);
    double pref = sqrt(dfact(l + m) * dfact(l - m) * dfact(l + nn) * dfact(l - nn));
    int k0 = (m - nn > 0) ? (m - nn) : 0;
    int k1 = (l + m < l - nn) ? (l + m) : (l - nn);
    double acc = 0.0;
    for (int k = k0; k <= k1; k++) {
      double den = dfact(l + m - k) * dfact(k) * dfact(l - nn - k) * dfact(nn - m + k);
      double term = ((k & 1) ? -1.0 : 1.0) / den
                  * dipow(cb, 2 * l + m - nn - 2 * k) * dipow(sb, nn - m + 2 * k);
      acc += term;
    }
    D[t] = (float)(pref * acc);
  }
}

// ---------------- init: quadrature weights ----------------------------------
__global__ void k_init_w(float* __restrict__ W) {
  int t = threadIdx.x;
  if (t >= W_TOTAL) return;
  const int DBv[NB] = {13, 10, 8, 5, 3, 2};
  const int WOFFv[NB] = {0, 26, 46, 62, 72, 78};
  int bi = 0;
  for (int i = 0; i < NB; i++)
    if (t >= WOFFv[i] && t < WOFFv[i] + 2 * DBv[i]) bi = i;
  int b = DBv[bi], j = t - WOFFv[bi];
  double beta = PI_D * (2.0 * j + 1.0) / (4.0 * b);
  double s = 0.0;
  for (int k = 0; k < b; k++) s += sin(beta * (2 * k + 1)) / (2 * k + 1);
  W[t] = (float)(2.0 / b * sin(beta) * s);
}

// ---------------- BN over (batch, spatial) per channel -----------------------
__global__ void k_bn0(const float* __restrict__ x, const float* __restrict__ g,
                      const float* __restrict__ bb, float* __restrict__ out) {
  const int C = 34, S = 26 * 26, Bn = 64;
  int c = blockIdx.x, tid = threadIdx.x;
  __shared__ float s1[256], s2[256];
  float sum = 0.f, sq = 0.f;
  for (int i = tid; i < Bn * S; i += 256) {
    int bi = i / S, si = i % S;
    float v = x[((long)bi * C + c) * S + si];
    sum += v; sq += v * v;
  }
  s1[tid] = sum; s2[tid] = sq; __syncthreads();
  for (int st = 128; st > 0; st >>= 1) {
    if (tid < st) { s1[tid] += s1[tid + st]; s2[tid] += s2[tid + st]; }
    __syncthreads();
  }
  float inv = 1.f / (float)(Bn * S);
  float mu = s1[0] * inv;
  float var = s2[0] * inv - mu * mu;
  float sc = g[c] * rsqrtf(var + 1e-5f);
  float sh = bb[c];
  for (int i = tid; i < Bn * S; i += 256) {
    int bi = i / S, si = i % S;
    long idx = ((long)bi * C + c) * S + si;
    out[idx] = sc * (x[idx] - mu) + sh;
  }
}

// ---------------- DFT passes (exact integer phase, LDS twiddles) ------------
__global__ void k_dft_last_real(const float* __restrict__ in, float* __restrict__ oRe,
                                float* __restrict__ oIm, long O, int N, float sign) {
  __shared__ float tc[32], ts[32];
  if (threadIdx.x < (unsigned)N) {
    float a = sign * 2.f * (float)PI_D * (float)threadIdx.x / (float)N;
    float s, c; sincosf(a, &s, &c); tc[threadIdx.x] = c; ts[threadIdx.x] = s;
  }
  __syncthreads();
  long t = (long)blockIdx.x * blockDim.x + threadIdx.x;
  if (t >= O * (long)N) return;
  long o = t / N; int m = (int)(t % N);
  const float* row = in + o * N;
  float re = 0.f, im = 0.f; int ph = 0;
  for (int g2 = 0; g2 < N; g2++) {
    float v = row[g2];
    re += v * tc[ph]; im += v * ts[ph];
    ph += m; if (ph >= N) ph -= N;
  }
  oRe[t] = re; oIm[t] = im;
}

__global__ void k_dft_last_c(const float* __restrict__ iRe, const float* __restrict__ iIm,
                             float* __restrict__ oRe, float* __restrict__ oIm,
                             long O, int N, float sign) {
  __shared__ float tc[32], ts[32];
  if (threadIdx.x < (unsigned)N) {
    float a = sign * 2.f * (float)PI_D * (float)threadIdx.x / (float)N;
    float s, c; sincosf(a, &s, &c); tc[threadIdx.x] = c; ts[threadIdx.x] = s;
  }
  __syncthreads();
  long t = (long)blockIdx.x * blockDim.x + threadIdx.x;
  if (t >= O * (long)N) return;
  long o = t / N; int m = (int)(t % N);
  const float* rr = iRe + o * N; const float* ri = iIm + o * N;
  float re = 0.f, im = 0.f; int ph = 0;
  for (int g2 = 0; g2 < N; g2++) {
    float vr = rr[g2], vi = ri[g2];
    re += vr * tc[ph] - vi * ts[ph];
    im += vr * ts[ph] + vi * tc[ph];
    ph += m; if (ph >= N) ph -= N;
  }
  oRe[t] = re; oIm[t] = im;
}

__global__ void k_dft_mid_c(const float* __restrict__ iRe, const float* __restrict__ iIm,
                            float* __restrict__ oRe, float* __restrict__ oIm,
                            long O, int N, int I, float sign) {
  __shared__ float tc[32], ts[32];
  if (threadIdx.x < (unsigned)N) {
    float a = sign * 2.f * (float)PI_D * (float)threadIdx.x / (float)N;
    float s, c; sincosf(a, &s, &c); tc[threadIdx.x] = c; ts[threadIdx.x] = s;
  }
  __syncthreads();
  long t = (long)blockIdx.x * blockDim.x + threadIdx.x;
  long total = O * (long)N * I;
  if (t >= total) return;
  long o = t / ((long)N * I);
  long r = t % ((long)N * I);
  int m = (int)(r / I), i = (int)(r % I);
  long base = (o * N) * (long)I + i;
  float re = 0.f, im = 0.f; int ph = 0;
  for (int a = 0; a < N; a++) {
    float vr = iRe[base + (long)a * I], vi = iIm[base + (long)a * I];
    re += vr * tc[ph] - vi * ts[ph];
    im += vr * ts[ph] + vi * tc[ph];
    ph += m; if (ph >= N) ph -= N;
  }
  oRe[t] = re; oIm[t] = im;
}

__global__ void k_dft_mid_real_relu(const float* __restrict__ iRe, const float* __restrict__ iIm,
                                    float* __restrict__ out, long O, int N, int I, float sign) {
  __shared__ float tc[32], ts[32];
  if (threadIdx.x < (unsigned)N) {
    float a = sign * 2.f * (float)PI_D * (float)threadIdx.x / (float)N;
    float s, c; sincosf(a, &s, &c); tc[threadIdx.x] = c; ts[threadIdx.x] = s;
  }
  __syncthreads();
  long t = (long)blockIdx.x * blockDim.x + threadIdx.x;
  long total = O * (long)N * I;
  if (t >= total) return;
  long o = t / ((long)N * I);
  long r = t % ((long)N * I);
  int m = (int)(r / I), i = (int)(r % I);
  long base = (o * N) * (long)I + i;
  float re = 0.f; int ph = 0;
  for (int a = 0; a < N; a++) {
    float vr = iRe[base + (long)a * I], vi = iIm[base + (long)a * I];
    re += vr * tc[ph] - vi * ts[ph];
    ph += m; if (ph >= N) ph -= N;
  }
  out[t] = fmaxf(re, 0.f);
}

// ---------------- per-l spectral extraction (writes zero-padded A) ----------
// A layout: Mp x Kp row-major, rows = (b*n+mi), cols = f (s2) / f*n+ki (so3).
// Padding columns [Kv, Kp) are written with zeros so the GEMM needs no guards.
__global__ void k_xh_s2(const float* __restrict__ XfRe, const float* __restrict__ XfIm,
                        const float* __restrict__ Dl, const float* __restrict__ Wq,
                        float* __restrict__ ARe, float* __restrict__ AIm,
                        int l, int M, int Kp) {
  const int F = 34, J = 26, Nin = 26;
  int n = 2 * l + 1;
  long total = (long)M * Kp;
  long t = (long)blockIdx.x * blockDim.x + threadIdx.x;
  if (t >= total) return;
  int col = (int)(t % Kp);
  int row = (int)(t / Kp);
  if (col >= F) { ARe[t] = 0.f; AIm[t] = 0.f; return; }
  int mi = row % n, b = row / n, f = col;
  int gi = pmod(mi - l, Nin);
  float re = 0.f, im = 0.f;
  for (int j = 0; j < J; j++) {
    float d = Dl[(j * n + mi) * n + l];          // dl0 = D[:, :, l]
    float w = Wq[j] * d;
    long xi = (((long)(b * F + f)) * J + j) * Nin + gi;
    re += w * XfRe[xi]; im += w * XfIm[xi];
  }
  ARe[t] = re; AIm[t] = im;
}

__global__ void k_xh_so3(const float* __restrict__ XfRe, const float* __restrict__ XfIm,
                         const float* __restrict__ Dl, const float* __restrict__ Wq,
                         float* __restrict__ ARe, float* __restrict__ AIm,
                         int F, int J, int Nin, int l, int M, int Kp) {
  int n = 2 * l + 1;
  int Kv = F * n;
  long total = (long)M * Kp;
  long t = (long)blockIdx.x * blockDim.x + threadIdx.x;
  if (t >= total) return;
  int col = (int)(t % Kp);
  int row = (int)(t / Kp);
  if (col >= Kv) { ARe[t] = 0.f; AIm[t] = 0.f; return; }
  int mi = row % n, b = row / n;
  int f = col / n, ki = col % n;
  int gm = pmod(mi - l, Nin), gk = pmod(ki - l, Nin);
  float re = 0.f, im = 0.f;
  for (int j = 0; j < J; j++) {
    float d = Dl[(j * n + mi) * n + ki];
    float w = Wq[j] * d;
    long xi = ((((long)(b * F + f)) * J + j) * Nin + gm) * Nin + gk;
    re += w * XfRe[xi]; im += w * XfIm[xi];
  }
  ARe[t] = re; AIm[t] = im;
}

// ---------------- weight packing: Bpk (Np x Kp, zero-padded) ----------------
__global__ void k_pack_s2(const float* __restrict__ w, float* __restrict__ Bp,
                          int F, int G, int WT, int off, int n, int Np, int Kp) {
  long total = (long)Np * Kp;
  long t = (long)blockIdx.x * blockDim.x + threadIdx.x;
  if (t >= total) return;
  int k = (int)(t % Kp);
  int rowN = (int)(t / Kp);
  if (rowN >= G * n || k >= F) { Bp[t] = 0.f; return; }
  int ni = rowN % n, g = rowN / n;
  Bp[t] = w[((long)k * G + g) * WT + off + ni];
}

__global__ void k_pack_so3(const float* __restrict__ w, float* __restrict__ Bp,
                           int F, int G, int WT, int off, int n, int Np, int Kp) {
  int Kv = F * n;
  long total = (long)Np * Kp;
  long t = (long)blockIdx.x * blockDim.x + threadIdx.x;
  if (t >= total) return;
  int k = (int)(t % Kp);
  int rowN = (int)(t / Kp);
  if (rowN >= G * n || k >= Kv) { Bp[t] = 0.f; return; }
  int nn = rowN % n, g = rowN / n;
  int f = k / n, kk = k % n;
  Bp[t] = w[((long)f * G + g) * WT + off + nn * n + kk];
}

// ---------------- WMMA f32 16x16x4 GEMM: C = A * B^T (all dims padded) -----
// A: Mp x Kp row-major.  Bm: Np x Kp row-major.  C: Mp x Np.
// Mp % 16 == 0, Np % 64 == 0, Kp % 4 == 0  ->  no guards, no EXEC churn.
__global__ void k_gemm_wmma(const float* __restrict__ A, const float* __restrict__ Bm,
                            float* __restrict__ C, int Mp, int Np, int Kp,
                            long aPlane, long cPlane) {
  int zb = blockIdx.z;
  const float* Ab = A + (long)zb * aPlane;
  float* Cb = C + (long)zb * cPlane;
  int wave = threadIdx.x >> 5;
  int lane = threadIdx.x & 31;
  int tm = blockIdx.y * 16;
  int tn = blockIdx.x * 64 + wave * 16;
  int lm = lane & 15;            // M index within tile (A) / N index (B)
  int lk = (lane >> 4) * 2;      // K base for this half-wave
  const float* arow = Ab + (long)(tm + lm) * Kp + lk;
  const float* brow = Bm + (long)(tn + lm) * Kp + lk;
  v8f acc = {};
  for (int k0 = 0; k0 < Kp; k0 += 4) {
    v2f a = *(const v2f*)(arow + k0);
    v2f b = *(const v2f*)(brow + k0);
    acc = __builtin_amdgcn_wmma_f32_16x16x4_f32(false, a, false, b, (short)0, acc, false, false);
  }
  int cn = tn + lm;
  int half = lane >> 4;
  float* ccol = Cb + (long)(tm + half * 8) * Np + cn;
  for (int r = 0; r < 8; r++) ccol[(long)r * Np] = acc[r];
}

// ---------------- accumulate z into SO(3) spectrum --------------------------
__global__ void k_accum(const float* __restrict__ zRe, const float* __restrict__ zIm,
                        const float* __restrict__ Dl, float* __restrict__ oRe,
                        float* __restrict__ oIm, int B, int G, int Jout, int Nout,
                        int l, int Ncols) {
  int n = 2 * l + 1;
  long total = (long)B * G * Jout * n * n;
  long t = (long)blockIdx.x * blockDim.x + threadIdx.x;
  if (t >= total) return;
  int ni = (int)(t % n);
  long r = t / n;
  int mi = (int)(r % n); r /= n;
  int j  = (int)(r % Jout); r /= Jout;
  int g  = (int)(r % G);
  int b  = (int)(r / G);
  float d = (float)(2 * l + 1) * Dl[(j * n + mi) * n + ni];
  long zi = ((long)(b * n + mi)) * Ncols + (long)g * n + ni;
  int im_ = pmod(mi - l, Nout), in_ = pmod(ni - l, Nout);
  long oi = ((((long)b * G + g) * Jout + j) * (long)Nout + im_) * Nout + in_;
  oRe[oi] += d * zRe[zi];
  oIm[oi] += d * zIm[zi];
}

// ---------------- integrate + FC head ---------------------------------------
__global__ void k_integrate(const float* __restrict__ x, const float* __restrict__ W2,
                            float* __restrict__ out) {
  long t = (long)blockIdx.x * blockDim.x + threadIdx.x;
  if (t >= 64L * 256) return;
  const float* base = x + t * 64;  // (j,a,g) 4x4x4
  float s = 0.f;
  for (int j = 0; j < 4; j++) {
    float wj = W2[j];
    for (int ag = 0; ag < 16; ag++) s += wj * base[j * 16 + ag];
  }
  out[t] = s * (1.f / 16.f);
}

__global__ void k_lin(const float* __restrict__ in, const float* __restrict__ W,
                      const float* __restrict__ bias, float* __restrict__ out,
                      int R, int I, int O) {
  long t = (long)blockIdx.x * blockDim.x + threadIdx.x;
  if (t >= (long)R * O) return;
  int r = (int)(t / O), o = (int)(t % O);
  float acc = bias[o];
  const float* xr = in + (long)r * I;
  const float* wr = W + (long)o * I;
  for (int i = 0; i < I; i++) acc += xr[i] * wr[i];
  out[t] = acc;
}

__global__ void k_bn_feat(float* __restrict__ xio, const float* __restrict__ g,
                          const float* __restrict__ b, int R, int F, int relu) {
  int f = blockIdx.x * blockDim.x + threadIdx.x;
  if (f >= F) return;
  float s = 0.f;
  for (int r = 0; r < R; r++) s += xio[(long)r * F + f];
  float mu = s / R;
  float v = 0.f;
  for (int r = 0; r < R; r++) { float d = xio[(long)r * F + f] - mu; v += d * d; }
  float sc = g[f] * rsqrtf(v / R + 1e-5f);
  for (int r = 0; r < R; r++) {
    float y = sc * (xio[(long)r * F + f] - mu) + b[f];
    if (relu) y = fmaxf(y, 0.f);
    xio[(long)r * F + f] = y;
  }
}

__global__ void k_final(const float* __restrict__ h, const float* __restrict__ w,
                        const float* __restrict__ bias, const float* __restrict__ g,
                        const float* __restrict__ bb, float* __restrict__ out) {
  __shared__ float s1[64], s2[64];
  int r = threadIdx.x;
  float t = bias[0];
  for (int i = 0; i < 32; i++) t += h[r * 32 + i] * w[i];
  s1[r] = t; s2[r] = t * t; __syncthreads();
  for (int st = 32; st > 0; st >>= 1) {
    if (r < st) { s1[r] += s1[r + st]; s2[r] += s2[r + st]; }
    __syncthreads();
  }
  float mu = s1[0] / 64.f;
  float var = s2[0] / 64.f - mu * mu;
  float y = g[0] * (t - mu) * rsqrtf(var + 1e-5f) + bb[0];
  out[r] = 1.f / (1.f + expf(-y));
}

// ---------------- host driver -----------------------------------------------
static inline unsigned gblocks(long n) { return (unsigned)((n + 255) / 256); }

extern "C" void kernel_launch(void* const* d_in, const int* in_sizes, int n_in,
                              void* d_out, int out_size, void* d_ws, size_t ws_size,
                              hipStream_t stream) {
  (void)in_sizes; (void)n_in; (void)out_size; (void)ws_size;
  const float* x = (const float*)d_in[0];
  const float* wc[5]  = {(const float*)d_in[1], (const float*)d_in[2], (const float*)d_in[3],
                         (const float*)d_in[4], (const float*)d_in[5]};
  const float* fcw[6] = {(const float*)d_in[6],  (const float*)d_in[8],  (const float*)d_in[10],
                         (const float*)d_in[12], (const float*)d_in[14], (const float*)d_in[16]};
  const float* fcb[6] = {(const float*)d_in[7],  (const float*)d_in[9],  (const float*)d_in[11],
                         (const float*)d_in[13], (const float*)d_in[15], (const float*)d_in[17]};
  const float* bng[7] = {(const float*)d_in[18], (const float*)d_in[20], (const float*)d_in[22],
                         (const float*)d_in[24], (const float*)d_in[26], (const float*)d_in[28],
                         (const float*)d_in[30]};
  const float* bnb[7] = {(const float*)d_in[19], (const float*)d_in[21], (const float*)d_in[23],
                         (const float*)d_in[25], (const float*)d_in[27], (const float*)d_in[29],
                         (const float*)d_in[31]};

  float* WS = (float*)d_ws;
  float* Dtab = WS + OFF_D;
  float* Wtab = WS + OFF_W;
  float* RA = WS + OFF_REAL;
  float* CA = WS + OFF_CA;
  float* CB = WS + OFF_CB;
  float* CC = WS + OFF_CC;
  float* AB = WS + OFF_A;
  float* BB = WS + OFF_B;
  float* ZB = WS + OFF_Z;
  float* FA = WS + OFF_FCA;
  float* FB = WS + OFF_FCB;

  k_init_d<<<512, 256, 0, stream>>>(Dtab);
  k_init_w<<<1, 128, 0, stream>>>(Wtab);
  k_bn0<<<34, 256, 0, stream>>>(x, bng[0], bnb[0], RA);

  struct Lr { int bin, bout, B, F, G, WT, lmax, din, dout; bool s2; };
  const Lr Ls[5] = {
    {13, 10, 64,  34,  64, 100, 10, 0, 1, true},
    {10,  8, 64,  64, 100, 680,  8, 1, 2, false},
    { 8,  5, 64, 100, 150, 165,  5, 2, 3, false},
    { 5,  3, 64, 150, 200,  35,  3, 3, 4, false},
    { 3,  2, 64, 200, 256,  10,  2, 4, 5, false},
  };

  for (int li = 0; li < 5; li++) {
    const Lr& P = Ls[li];
    int Nin = 2 * P.bin, J = Nin, Jout = 2 * P.bout, Nout = 2 * P.bout;
    const float* w = wc[li];

    if (P.s2) {
      long O = (long)P.B * P.F * Nin;  // 1D fft over gamma only
      k_dft_last_real<<<gblocks(O * Nin), 256, 0, stream>>>(RA, CA, CA + PLANE, O, Nin, -1.f);
    } else {
      long O1 = (long)P.B * P.F * J * Nin;
      k_dft_last_real<<<gblocks(O1 * Nin), 256, 0, stream>>>(RA, CC, CC + PLANE, O1, Nin, -1.f);
      long O2 = (long)P.B * P.F * J;
      k_dft_mid_c<<<gblocks(O2 * (long)Nin * Nin), 256, 0, stream>>>(
          CC, CC + PLANE, CA, CA + PLANE, O2, Nin, Nin, -1.f);
    }

    long outSz = (long)P.B * P.G * Jout * Nout * Nout;
    hipMemsetAsync((void*)CB, 0, (size_t)outSz * sizeof(float), stream);
    hipMemsetAsync((void*)(CB + PLANE), 0, (size_t)outSz * sizeof(float), stream);

    int off = 0;
    for (int l = 0; l < P.lmax; l++) {
      int n = 2 * l + 1;
      const float* Din  = Dtab + d_off(P.din, l);
      const float* Dout = Dtab + d_off(P.dout, l);
      const float* Wq   = Wtab + WOFFh[P.din];
      int M = P.B * n;                       // multiple of 64 already
      int N = P.G * n;
      int K = P.s2 ? P.F : P.F * n;
      int Np = (N + 63) & ~63;
      int Kp = (K + 3) & ~3;
      if (P.s2) {
        k_xh_s2<<<gblocks((long)M * Kp), 256, 0, stream>>>(
            CA, CA + PLANE, Din, Wq, AB, AB + APL, l, M, Kp);
        k_pack_s2<<<gblocks((long)Np * Kp), 256, 0, stream>>>(w, BB, P.F, P.G, P.WT, off, n, Np, Kp);
        off += n;
      } else {
        k_xh_so3<<<gblocks((long)M * Kp), 256, 0, stream>>>(
            CA, CA + PLANE, Din, Wq, AB, AB + APL, P.F, J, Nin, l, M, Kp);
        k_pack_so3<<<gblocks((long)Np * Kp), 256, 0, stream>>>(w, BB, P.F, P.G, P.WT, off, n, Np, Kp);
        off += n * n;
      }
      dim3 gg((unsigned)(Np / 64), (unsigned)(M / 16), 2);
      k_gemm_wmma<<<gg, 128, 0, stream>>>(AB, BB, ZB, M, Np, Kp, APL, ZPL);
      k_accum<<<gblocks((long)P.B * P.G * Jout * n * n), 256, 0, stream>>>(
          ZB, ZB + ZPL, Dout, CB, CB + PLANE, P.B, P.G, Jout, Nout, l, Np);
    }

    long O1 = (long)P.B * P.G * Jout * Nout;
    k_dft_last_c<<<gblocks(O1 * Nout), 256, 0, stream>>>(CB, CB + PLANE, CC, CC + PLANE, O1, Nout, 1.f);
    long O2 = (long)P.B * P.G * Jout;
    k_dft_mid_real_relu<<<gblocks(O2 * (long)Nout * Nout), 256, 0, stream>>>(
        CC, CC + PLANE, RA, O2, Nout, Nout, 1.f);
  }

  k_integrate<<<gblocks(64L * 256), 256, 0, stream>>>(RA, Wtab + WOFFh[5], FA);

  k_lin<<<gblocks(64L * 512), 256, 0, stream>>>(FA, fcw[0], fcb[0], FB, 64, 256, 512);
  k_bn_feat<<<gblocks(512), 256, 0, stream>>>(FB, bng[1], bnb[1], 64, 512, 1);
  k_lin<<<gblocks(64L * 256), 256, 0, stream>>>(FB, fcw[1], fcb[1], FA, 64, 512, 256);
  k_bn_feat<<<gblocks(256), 256, 0, stream>>>(FA, bng[2], bnb[2], 64, 256, 1);
  k_lin<<<gblocks(64L * 128), 256, 0, stream>>>(FA, fcw[2], fcb[2], FB, 64, 256, 128);
  k_bn_feat<<<gblocks(128), 256, 0, stream>>>(FB, bng[3], bnb[3], 64, 128, 1);
  k_lin<<<gblocks(64L * 64), 256, 0, stream>>>(FB, fcw[3], fcb[3], FA, 64, 128, 64);
  k_bn_feat<<<gblocks(64), 256, 0, stream>>>(FA, bng[4], bnb[4], 64, 64, 1);
  k_lin<<<gblocks(64L * 32), 256, 0, stream>>>(FA, fcw[4], fcb[4], FB, 64, 64, 32);
  k_bn_feat<<<gblocks(32), 256, 0, stream>>>(FB, bng[5], bnb[5], 64, 32, 1);
  k_final<<<1, 64, 0, stream>>>(FB, fcw[5], fcb[5], bng[6], bnb[6], (float*)d_out);
}